// DynamicTanh_14611478741530
// MI455X (gfx1250) — compile-verified
//
#include <hip/hip_runtime.h>
#include <math.h>

#define BATCH 64
#define TLEN  4096
#define FDIM  256
#define KDIM  512            // 2*F
#define NCHUNK 32            // reduction chunks per batch
#define CHUNK_T (TLEN / NCHUNK)

typedef __attribute__((ext_vector_type(2))) float v2f;
typedef __attribute__((ext_vector_type(8))) float v8f;

// ---------------------------------------------------------------------------
// Pass 1a: partial sums over T. grid = (NCHUNK, BATCH), block = 256 (one
// thread per feature). Coalesced: 32 lanes read 128B contiguous per step.
// ---------------------------------------------------------------------------
__global__ void reduce_partial_kernel(const float* __restrict__ x,
                                      float* __restrict__ partial) {
  const int b = blockIdx.y;
  const int c = blockIdx.x;
  const int f = threadIdx.x;
  const float* xp = x + ((size_t)b * TLEN + (size_t)c * CHUNK_T) * FDIM + f;
  float s = 0.f, s2 = 0.f;
#pragma unroll 4
  for (int t = 0; t < CHUNK_T; ++t) {
    float v = xp[(size_t)t * FDIM];
    s  += v;
    s2 += v * v;
  }
  float* p = partial + ((size_t)(b * NCHUNK + c) * 2) * FDIM;
  p[f]        = s;
  p[FDIM + f] = s2;
}

// ---------------------------------------------------------------------------
// Pass 1b: finalize mean / std into stats[b, 0:256]=mean, stats[b,256:512]=std
// grid = BATCH, block = 256. Fixed summation order -> deterministic.
// ---------------------------------------------------------------------------
__global__ void reduce_final_kernel(const float* __restrict__ partial,
                                    float* __restrict__ stats) {
  const int b = blockIdx.x;
  const int f = threadIdx.x;
  float s = 0.f, s2 = 0.f;
  for (int c = 0; c < NCHUNK; ++c) {
    const float* p = partial + ((size_t)(b * NCHUNK + c) * 2) * FDIM;
    s  += p[f];
    s2 += p[FDIM + f];
  }
  const float inv = 1.0f / (float)TLEN;
  float mean = s * inv;
  float var  = s2 * inv - mean * mean;
  float sd   = sqrtf(var + 1e-5f);
  stats[b * KDIM + f]        = mean;
  stats[b * KDIM + FDIM + f] = sd;
}

// ---------------------------------------------------------------------------
// Pass 2: MLP gates via fp32 WMMA (V_WMMA_F32_16X16X4_F32).
//   H     = relu(stats[64,512] @ W1^T + b1)        (GEMM1, K=512)
//   alpha = sigmoid(H[64,256] @ W2^T + b2)         (GEMM2, K=256)
//   scale[b,f] = alpha / std                       (folds rstd for pass 3)
// Single workgroup: 1024 threads = 32 waves; 64 16x16 tiles per GEMM,
// 2 tiles/wave. H lives in LDS (64 KB). EXEC all ones (WMMA requirement).
//
// Fragment layouts per CDNA5 ISA 7.12.2:
//   A 16x4  f32 (2 VGPR): lane m = lane&15, K base = (lane>>4)*2
//   B 4x16  f32 (2 VGPR): lane n = lane&15, K base = (lane>>4)*2
//   C/D 16x16 f32 (8 VGPR): elem v -> M = v + 8*(lane>>4), N = lane&15
// ---------------------------------------------------------------------------
__global__ void mlp_wmma_kernel(const float* __restrict__ stats,  // [64,512]
                                const float* __restrict__ W1,     // [256,512]
                                const float* __restrict__ b1,     // [256]
                                const float* __restrict__ W2,     // [256,256]
                                const float* __restrict__ b2,     // [256]
                                float* __restrict__ scale) {      // [64,256]
  __shared__ float Hs[BATCH * FDIM];  // 64 KB

  const int tid  = threadIdx.x;
  const int wave = tid >> 5;
  const int lane = tid & 31;
  const int m    = lane & 15;          // row (A) / col (B) within tile
  const int kb   = (lane >> 4) * 2;    // K sub-offset: 0 or 2
  const int mv   = (lane >> 4) * 8;    // M offset for C/D elements

  // ---- GEMM1: H = relu(stats @ W1^T + b1) ----
  for (int tile = wave; tile < 64; tile += 32) {
    const int mt = tile >> 4;          // 0..3  (batch tile)
    const int nt = tile & 15;          // 0..15 (feature tile)
    const int arow = mt * 16 + m;      // batch row for A
    const int bcol = nt * 16 + m;      // output feature (row of W1)
    v8f acc = {};
    for (int k0 = 0; k0 < KDIM; k0 += 4) {
      v2f a, b;
      a.x = stats[arow * KDIM + k0 + kb];
      a.y = stats[arow * KDIM + k0 + kb + 1];
      // B[k][n] = W1^T[k][n] = W1[n][k]
      b.x = W1[bcol * KDIM + k0 + kb];
      b.y = W1[bcol * KDIM + k0 + kb + 1];
      acc = __builtin_amdgcn_wmma_f32_16x16x4_f32(false, a, false, b,
                                                  (short)0, acc, false, false);
    }
    const int nOut = nt * 16 + (lane & 15);
    const float bias = b1[nOut];
#pragma unroll
    for (int v = 0; v < 8; ++v) {
      const int mOut = mt * 16 + mv + v;
      float h = acc[v] + bias;
      Hs[mOut * FDIM + nOut] = h > 0.f ? h : 0.f;
    }
  }
  __syncthreads();

  // ---- GEMM2: alpha = sigmoid(H @ W2^T + b2); scale = alpha / std ----
  for (int tile = wave; tile < 64; tile += 32) {
    const int mt = tile >> 4;
    const int nt = tile & 15;
    const int arow = mt * 16 + m;
    const int bcol = nt * 16 + m;
    v8f acc = {};
    for (int k0 = 0; k0 < FDIM; k0 += 4) {
      v2f a, b;
      a.x = Hs[arow * FDIM + k0 + kb];
      a.y = Hs[arow * FDIM + k0 + kb + 1];
      b.x = W2[bcol * FDIM + k0 + kb];
      b.y = W2[bcol * FDIM + k0 + kb + 1];
      acc = __builtin_amdgcn_wmma_f32_16x16x4_f32(false, a, false, b,
                                                  (short)0, acc, false, false);
    }
    const int nOut = nt * 16 + (lane & 15);
    const float bias = b2[nOut];
#pragma unroll
    for (int v = 0; v < 8; ++v) {
      const int mOut = mt * 16 + mv + v;           // batch index
      float z = acc[v] + bias;
      float alpha = 1.0f / (1.0f + __expf(-z));
      float sd = stats[mOut * KDIM + FDIM + nOut]; // std[b,f]
      scale[mOut * FDIM + nOut] = alpha / sd;
    }
  }
}

// ---------------------------------------------------------------------------
// Pass 3: out = gamma * tanh(scale * (x - mean)) + beta.  float4 vectorized.
// grid = (TLEN/64, BATCH), block = 256. Each thread owns 4 features, loops
// over 16 timesteps; per-feature params hoisted into registers.
// ---------------------------------------------------------------------------
__global__ void apply_kernel(const float* __restrict__ x,
                             const float* __restrict__ stats,
                             const float* __restrict__ scale,
                             const float* __restrict__ gamma,
                             const float* __restrict__ beta,
                             float* __restrict__ out) {
  const int b   = blockIdx.y;
  const int tid = threadIdx.x;
  const int f4  = (tid & 63) * 4;
  const int t0  = blockIdx.x * 64;

  const float4 mean4 = *(const float4*)(stats + b * KDIM + f4);
  const float4 sc4   = *(const float4*)(scale + b * FDIM + f4);
  const float4 g4    = *(const float4*)(gamma + f4);
  const float4 be4   = *(const float4*)(beta + f4);

  const size_t base = (size_t)b * TLEN * FDIM;
#pragma unroll 4
  for (int t = t0 + (tid >> 6); t < t0 + 64; t += 4) {
    const float4 xv = *(const float4*)(x + base + (size_t)t * FDIM + f4);
    float4 o;
    o.x = g4.x * tanhf(sc4.x * (xv.x - mean4.x)) + be4.x;
    o.y = g4.y * tanhf(sc4.y * (xv.y - mean4.y)) + be4.y;
    o.z = g4.z * tanhf(sc4.z * (xv.z - mean4.z)) + be4.z;
    o.w = g4.w * tanhf(sc4.w * (xv.w - mean4.w)) + be4.w;
    *(float4*)(out + base + (size_t)t * FDIM + f4) = o;
  }
}

// ---------------------------------------------------------------------------
extern "C" void kernel_launch(void* const* d_in, const int* in_sizes, int n_in,
                              void* d_out, int out_size, void* d_ws, size_t ws_size,
                              hipStream_t stream) {
  const float* x     = (const float*)d_in[0];
  const float* gamma = (const float*)d_in[1];
  const float* beta  = (const float*)d_in[2];
  const float* W1    = (const float*)d_in[3];
  const float* b1    = (const float*)d_in[4];
  const float* W2    = (const float*)d_in[5];
  const float* b2    = (const float*)d_in[6];
  float* out = (float*)d_out;

  // Workspace layout (floats):
  float* partial = (float*)d_ws;                         // 64*32*2*256 = 1,048,576
  float* stats   = partial + (size_t)BATCH * NCHUNK * 2 * FDIM;  // 64*512
  float* scale   = stats + (size_t)BATCH * KDIM;                 // 64*256
  // total ~4.2 MB

  reduce_partial_kernel<<<dim3(NCHUNK, BATCH), 256, 0, stream>>>(x, partial);
  reduce_final_kernel<<<BATCH, 256, 0, stream>>>(partial, stats);
  mlp_wmma_kernel<<<1, 1024, 0, stream>>>(stats, W1, b1, W2, b2, scale);
  apply_kernel<<<dim3(TLEN / 64, BATCH), 256, 0, stream>>>(x, stats, scale,
                                                           gamma, beta, out);
}